// RoIModule_85469849190576
// MI455X (gfx1250) — compile-verified
//
#include <hip/hip_runtime.h>
#include <stdint.h>

namespace {
constexpr int kBatch = 64;
constexpr int kNBox  = 16 * 64 * 64;   // 65536 boxes per image
constexpr int kRSel  = 128;
constexpr int kThr   = 1024;           // 32 wave32 waves -> one WGP per image
constexpr int kKpt   = kNBox / kThr;   // 64 boxes/scores per thread (VGPR-resident scores)
constexpr int kKLds  = 16;             // first 16*1024 boxes LDS-resident (256 KB of 320 KB)
constexpr float kNmsT = 0.3f;
constexpr float kPen  = 5.0f;
}

typedef unsigned int u32x4 __attribute__((ext_vector_type(4)));
typedef int i32x8 __attribute__((ext_vector_type(8)));
typedef int i32x4 __attribute__((ext_vector_type(4)));

__global__ __launch_bounds__(kThr, 1)
void nms_select_kernel(const float* __restrict__ scores,
                       const float* __restrict__ boxes,
                       float* __restrict__ out) {
  const int img  = blockIdx.x;
  const int tid  = threadIdx.x;
  const int lane = tid & 31;          // wave32
  const int wid  = tid >> 5;

  const float*  sc = scores + (size_t)img * kNBox;
  const float4* bx = reinterpret_cast<const float4*>(boxes) + (size_t)img * kNBox;

  __shared__ float4 ldsBoxes[kKLds * kThr];   // 256 KB box prefix, resident all 128 rounds
  __shared__ float  redV[32];
  __shared__ int    redI[32];
  __shared__ float  bc[5];                    // winner x1,y1,x2,y2,area broadcast

  // ---- One-shot TDM copy: first 16384 boxes (2D tile: 4096 floats x 16 rows) ----
#if __has_builtin(__builtin_amdgcn_tensor_load_to_lds)
  if (wid == 0) {
    const uint64_t ga = (uint64_t)(uintptr_t)bx;
    // flat LDS pointer: low 32 bits == byte offset within the workgroup's LDS
    const uint32_t la =
        __builtin_amdgcn_readfirstlane((uint32_t)(uintptr_t)&ldsBoxes[0]);
    const uint32_t galo = __builtin_amdgcn_readfirstlane((uint32_t)ga);
    const uint32_t gahi = __builtin_amdgcn_readfirstlane((uint32_t)(ga >> 32));

    constexpr uint32_t td0 = 4096, td1 = 16, tile0 = 4096, tile1 = 16;
    constexpr uint64_t str0 = 4096;                 // dim0 stride (elements)
    constexpr uint64_t str1 = (uint64_t)td0 * td1;  // dim1 stride (unused, tile_dim2=0)

    u32x4 g0;
    g0[0] = 1u;                                   // count=1 valid descriptor, user mode
    g0[1] = la;                                   // lds_addr (bytes)
    g0[2] = galo;                                 // global_addr[31:0]
    g0[3] = (gahi & 0x01FFFFFFu) | 0x80000000u;   // global_addr[56:32] | type=2

    i32x8 g1;
    g1[0] = (int)(2u << 16);                      // data_size=2 -> 4-byte elements
    g1[1] = (int)((td0 & 0xFFFFu) << 16);                               // tensor_dim0 lo16
    g1[2] = (int)(((td0 >> 16) & 0xFFFFu) | ((td1 & 0xFFFFu) << 16));   // dim0 hi | dim1 lo
    g1[3] = (int)(((td1 >> 16) & 0xFFFFu) | ((tile0 & 0xFFFFu) << 16)); // dim1 hi | tile0
    g1[4] = (int)(tile1 & 0xFFFFu);                                     // tile1 | tile2=0
    g1[5] = (int)(uint32_t)str0;                                        // stride0 lo32
    g1[6] = (int)(((uint32_t)(str0 >> 32) & 0xFFFFu) |
                  (uint32_t)((str1 & 0xFFFFu) << 16));                  // s0 hi16 | s1 lo16
    g1[7] = (int)(uint32_t)(str1 >> 16);                                // stride1 hi32
    i32x4 gz4 = {0, 0, 0, 0};                     // 2D tensor: groups 2/3 zero
    i32x8 gz8 = {0, 0, 0, 0, 0, 0, 0, 0};        // clang-23 lane: extra 6th group arg
    __builtin_amdgcn_tensor_load_to_lds(g0, g1, gz4, gz4, gz8, 0);
    __builtin_amdgcn_s_wait_tensorcnt(0);
  }
#else
  for (int i = tid; i < kKLds * kThr; i += kThr) ldsBoxes[i] = bx[i];
#endif

  // Scores into VGPRs (coalesced, layout j = k*1024 + tid)
  float s[kKpt];
#pragma unroll
  for (int k = 0; k < kKpt; ++k) s[k] = sc[k * kThr + tid];

  __syncthreads();   // LDS box prefix visible to all waves

  float4* outB = reinterpret_cast<float4*>(out);          // (B, R, 4)
  float*  outS = out + (size_t)kBatch * kRSel * 4;        // (B, R)

  for (int step = 0; step < kRSel; ++step) {
    // ---- 1. per-thread argmax over register scores (ascending j => min-index ties) ----
    float bv = -3.402823466e38f;
    int   bi = 0x7FFFFFFF;
#pragma unroll
    for (int k = 0; k < kKpt; ++k) {
      if (s[k] > bv) { bv = s[k]; bi = k * kThr + tid; }
    }
    // ---- 2. wave32 shuffle-xor reduction with lowest-index tie-break ----
#pragma unroll
    for (int off = 16; off > 0; off >>= 1) {
      float ov = __shfl_xor(bv, off, 32);
      int   oi = __shfl_xor(bi, off, 32);
      if (ov > bv || (ov == bv && oi < bi)) { bv = ov; bi = oi; }
    }
    if (lane == 0) { redV[wid] = bv; redI[wid] = bi; }
    __syncthreads();
    // ---- 3. cross-wave reduce (wave 0), winner fetch + reversed-order output ----
    if (wid == 0) {
      float v = redV[lane];
      int   i = redI[lane];
#pragma unroll
      for (int off = 16; off > 0; off >>= 1) {
        float ov = __shfl_xor(v, off, 32);
        int   oi = __shfl_xor(i, off, 32);
        if (ov > v || (ov == v && oi < i)) { v = ov; i = oi; }
      }
      if (lane == 0) {
        float4 wb = bx[i];
        float  wa = (wb.z - wb.x) * (wb.w - wb.y);
        bc[0] = wb.x; bc[1] = wb.y; bc[2] = wb.z; bc[3] = wb.w; bc[4] = wa;
        const int row = img * kRSel + (kRSel - 1 - step);  // inds[::-1]
        outB[row] = wb;
        outS[row] = sc[i];     // ORIGINAL score, not the suppressed one
      }
    }
    __syncthreads();
    // ---- 4. suppression sweep: LDS prefix + L2-resident tail, scores stay in VGPRs ----
    const float wx1 = bc[0], wy1 = bc[1], wx2 = bc[2], wy2 = bc[3], wa = bc[4];
#pragma unroll
    for (int k = 0; k < kKpt; ++k) {
      const int j = k * kThr + tid;
      float4 b = (k < kKLds) ? ldsBoxes[j] : bx[j];
      float dx = fminf(wx2, b.z) - fmaxf(wx1, b.x);
      float dy = fminf(wy2, b.w) - fmaxf(wy1, b.y);
      dx = fmaxf(dx, 0.0f);
      dy = fmaxf(dy, 0.0f);
      const float inter = dx * dy;
      const float area  = (b.z - b.x) * (b.w - b.y);
      const float denom = fmaxf(wa + area - inter, 1.0f);   // >= 1, so compare w/o divide
      if (inter > kNmsT * denom) s[k] -= kPen;              // iou > 0.3
    }
  }
}

extern "C" void kernel_launch(void* const* d_in, const int* in_sizes, int n_in,
                              void* d_out, int out_size, void* d_ws, size_t ws_size,
                              hipStream_t stream) {
  (void)in_sizes; (void)n_in; (void)out_size; (void)d_ws; (void)ws_size;
  const float* scores = (const float*)d_in[0];  // (B, A, G, G)
  const float* boxes  = (const float*)d_in[1];  // (B, A, G, G, 4)
  float* out = (float*)d_out;                   // (B,R,4) boxes then (B,R) scores
  nms_select_kernel<<<dim3(kBatch), dim3(kThr), 0, stream>>>(scores, boxes, out);
}